// GraphConv1_15470472200484
// MI455X (gfx1250) — compile-verified
//
#include <hip/hip_runtime.h>

typedef float v2f __attribute__((ext_vector_type(2)));
typedef float v4f __attribute__((ext_vector_type(4)));
typedef float v8f __attribute__((ext_vector_type(8)));

#define FEAT 32
#define BPAD 20  // per-column LDS row stride in floats (80 B: 16B-aligned, bank-spread)

// ---------------------------------------------------------------------------
// Kernel 1: zero the accumulation workspace (sums[N*32] + cnt[N]), float4.
// ---------------------------------------------------------------------------
__global__ void zero_ws_kernel(float* __restrict__ ws, int n) {
  int i = blockIdx.x * blockDim.x + threadIdx.x;
  int i4 = i * 4;
  if (i4 + 3 < n) {
    *(v4f*)(ws + i4) = (v4f){0.0f, 0.0f, 0.0f, 0.0f};
  } else {
    for (int k = i4; k < n; ++k) ws[k] = 0.0f;
  }
}

// ---------------------------------------------------------------------------
// Kernel 2: edge scatter.  One lane per (edge, feature).
// sums[src][f] += features[tgt][f];  cnt[src] += 1  (lane f==0 only).
// Relaxed agent-scope float atomics -> global_atomic_add_f32 (L2 RMW units).
// ---------------------------------------------------------------------------
__global__ void edge_scatter_kernel(const float* __restrict__ feat,
                                    const int* __restrict__ esrc,
                                    const int* __restrict__ etgt,
                                    float* __restrict__ sums,
                                    float* __restrict__ cnt,
                                    int num_edges) {
  long long idx = (long long)blockIdx.x * blockDim.x + threadIdx.x;
  int e = (int)(idx >> 5);
  int f = (int)(idx & 31);
  if (e >= num_edges) return;
  int s = esrc[e];
  int t = etgt[e];
  float v = feat[(long long)t * FEAT + f];
  __hip_atomic_fetch_add(&sums[(long long)s * FEAT + f], v,
                         __ATOMIC_RELAXED, __HIP_MEMORY_SCOPE_AGENT);
  if (f == 0) {
    __hip_atomic_fetch_add(&cnt[s], 1.0f,
                           __ATOMIC_RELAXED, __HIP_MEMORY_SCOPE_AGENT);
  }
}

// ---------------------------------------------------------------------------
// Kernel 3: fused dual GEMM + mean + relu + concat via V_WMMA_F32_16X16X4_F32.
// One wave per 16-node row tile, 8 waves (256 threads) per block.
//
// Weight is staged once per block into LDS in a k-pair-interleaved transposed
// layout:  wtp[h][col][p]  with p = kk*2 + j  holding  W[kk*4 + 2h + j][col].
// A lane (r = lane%16, h = lane/16) then reads its whole B operand stream
// (columns r and r+16, all K) as 8x ds_load_b128, fully hoisted out of the
// k-loop; the inner loop issues only the two A-operand global b64 loads and
// four WMMAs per K-step.
//
// f32 WMMA operand layouts (ISA 7.12.2):
//   A 16x4 : lane L -> row L%16, VGPR j -> K = 2*(L/16)+j
//   B 4x16 : lane L -> col L%16, VGPR j -> K = 2*(L/16)+j
//   C/D    : VGPR j -> row j + 8*(L/16), col L%16
// ---------------------------------------------------------------------------
__global__ void gcn_wmma_kernel(const float* __restrict__ feat,
                                const float* __restrict__ W,
                                const float* __restrict__ sums,
                                const float* __restrict__ cnt,
                                float* __restrict__ out,
                                int num_nodes) {
  __shared__ float wtp[2 * 32 * BPAD];  // [h][col][pair], 5.0 KB

  // ---- stage W into LDS (coalesced float4 reads, one per thread) ----------
  {
    const int tid = threadIdx.x;          // blockDim.x == 256
    v4f w4 = ((const v4f*)W)[tid];        // elements 4*tid .. 4*tid+3 (same k)
    const int k    = tid >> 3;            // 0..31
    const int colb = (tid & 7) << 2;      // 0,4,..,28
    const int kk   = k >> 2;
    const int rem  = k & 3;
    const int hh   = rem >> 1;
    const int j    = rem & 1;
    const int p    = kk * 2 + j;
    wtp[(hh * 32 + colb + 0) * BPAD + p] = w4[0];
    wtp[(hh * 32 + colb + 1) * BPAD + p] = w4[1];
    wtp[(hh * 32 + colb + 2) * BPAD + p] = w4[2];
    wtp[(hh * 32 + colb + 3) * BPAD + p] = w4[3];
  }
  __syncthreads();

  const int lane = threadIdx.x & 31;
  const int wave = threadIdx.x >> 5;
  const int tile = blockIdx.x * (blockDim.x >> 5) + wave;
  const int rowBase = tile * 16;
  if (rowBase >= num_nodes) return;  // wave-uniform: EXEC stays all-1s

  const int r = lane & 15;   // row (A) / col (B,C,D) within tile
  const int h = lane >> 4;   // half select for K / M
  const int row = rowBase + r;

  // mean divisor: single v_rcp_f32 instead of full IEEE divide chain
  const float invc = __builtin_amdgcn_rcpf(fmaxf(cnt[row], 1.0f));

  // ---- hoist all B operands: 16 floats per column set, 16B-aligned --------
  const float* b0p = &wtp[(h * 32 + r) * BPAD];
  const float* b1p = &wtp[(h * 32 + 16 + r) * BPAD];
  v4f B0[4], B1[4];
#pragma unroll
  for (int q = 0; q < 4; ++q) {
    B0[q] = *(const v4f*)(b0p + 4 * q);   // ds_load_b128
    B1[q] = *(const v4f*)(b1p + 4 * q);   // ds_load_b128
  }

  v8f acc0 = {};
  v8f acc1 = {};
  v8f acc2 = {};
  v8f acc3 = {};

#pragma unroll
  for (int kk = 0; kk < 8; ++kk) {
    const int k0 = kk * 4 + h * 2;

    // A operands: two contiguous f32 per lane (8-byte aligned)
    v2f aF = *(const v2f*)(feat + (long long)row * FEAT + k0);
    v2f aS = *(const v2f*)(sums + (long long)row * FEAT + k0);
    v2f aA;
    aA.x = aS.x * invc;
    aA.y = aS.y * invc;

    // B operands from the hoisted registers (compile-time indices)
    const int q = kk >> 1;
    const int c = (kk & 1) * 2;
    v2f b0; b0.x = B0[q][c]; b0.y = B0[q][c + 1];
    v2f b1; b1.x = B1[q][c]; b1.y = B1[q][c + 1];

    acc0 = __builtin_amdgcn_wmma_f32_16x16x4_f32(false, aF, false, b0,
                                                 (short)0, acc0, false, false);
    acc1 = __builtin_amdgcn_wmma_f32_16x16x4_f32(false, aF, false, b1,
                                                 (short)0, acc1, false, false);
    acc2 = __builtin_amdgcn_wmma_f32_16x16x4_f32(false, aA, false, b0,
                                                 (short)0, acc2, false, false);
    acc3 = __builtin_amdgcn_wmma_f32_16x16x4_f32(false, aA, false, b1,
                                                 (short)0, acc3, false, false);
  }

  // ---- ReLU + store into concatenated [N, 64] output ----------------------
#pragma unroll
  for (int j = 0; j < 8; ++j) {
    int node = rowBase + j + 8 * h;
    float* o = out + (long long)node * 64;
    o[r]      = fmaxf(acc0[j], 0.0f);
    o[16 + r] = fmaxf(acc1[j], 0.0f);
    o[32 + r] = fmaxf(acc2[j], 0.0f);
    o[48 + r] = fmaxf(acc3[j], 0.0f);
  }
}

// ---------------------------------------------------------------------------
extern "C" void kernel_launch(void* const* d_in, const int* in_sizes, int n_in,
                              void* d_out, int out_size, void* d_ws, size_t ws_size,
                              hipStream_t stream) {
  const float* feat = (const float*)d_in[0];   // [N, 32]
  const float* W    = (const float*)d_in[1];   // [32, 32]
  const int* esrc   = (const int*)d_in[2];     // [E]
  const int* etgt   = (const int*)d_in[3];     // [E]
  (void)n_in; (void)ws_size; (void)out_size;

  const int N = in_sizes[0] / FEAT;
  const int E = in_sizes[2];

  float* out  = (float*)d_out;                 // [N, 64]
  float* sums = (float*)d_ws;                  // [N, 32]
  float* cnt  = sums + (size_t)N * FEAT;       // [N]

  // 1) zero workspace (every call: harness does not re-poison between replays)
  const int zn = N * (FEAT + 1);
  const int zthreads = (zn + 3) / 4;
  zero_ws_kernel<<<(zthreads + 255) / 256, 256, 0, stream>>>((float*)d_ws, zn);

  // 2) edge scatter: E*32 lanes
  const long long tot = (long long)E * 32;
  const int sblocks = (int)((tot + 255) / 256);
  edge_scatter_kernel<<<sblocks, 256, 0, stream>>>(feat, esrc, etgt, sums, cnt, E);

  // 3) fused WMMA GEMMs + mean + relu + concat: one wave per 16 rows
  const int tiles = (N + 15) / 16;
  const int wavesPerBlock = 8;  // 256 threads = 8 x wave32
  const int gblocks = (tiles + wavesPerBlock - 1) / wavesPerBlock;
  gcn_wmma_kernel<<<gblocks, wavesPerBlock * 32, 0, stream>>>(feat, W, sums, cnt,
                                                              out, N);
}